// DPLR_CS_SSM_62302795596592
// MI455X (gfx1250) — compile-verified
//
#include <hip/hip_runtime.h>
#include <hip/hip_bf16.h>

#define N16    16
#define DIN    64
#define DOUT   64
#define BATCH  32
#define SEQ    8192
#define CHUNK  16
#define NCHUNK (SEQ / CHUNK)   // 512
#define LOG2_CHUNK 4

// workspace layout (float offsets)
#define WS_AD   0                              // 256   : A_d row-major [i][k]
#define WS_T    256                            // 256   : A_d^CHUNK
#define WS_BD   512                            // 1024  : B_d row-major [n][d]
#define WS_CT   1536                           // 1024  : C^T  [k][o]
#define WS_DT   2560                           // 4096  : D^T  [d][o]
#define WS_F    6656                           // NCHUNK*512 : local finals
#define WS_HIN  (WS_F + NCHUNK * BATCH * N16)  // NCHUNK*512 : chunk incoming states
#define WS_U    (WS_HIN + NCHUNK * BATCH * N16)// BATCH*SEQ*16 : U = B_d x

typedef float v2f __attribute__((ext_vector_type(2)));
typedef float v8f __attribute__((ext_vector_type(8)));

// ---------------------------------------------------------------------------
// K1: setup — A_c = P(diag(-exp(ll)) + p q^T)P^T ; Gauss-Jordan solve for
// A_d = (I-A_c/2)^{-1}(I+A_c/2), B_d = (I-A_c/2)^{-1} B ; T = A_d^CHUNK ;
// transposes of C and D for WMMA B-fragments.
// ---------------------------------------------------------------------------
__global__ __launch_bounds__(256) void ssm_setup(
    const float* __restrict__ log_lambda, const float* __restrict__ p,
    const float* __restrict__ q, const float* __restrict__ B,
    const float* __restrict__ C, const float* __restrict__ D,
    const float* __restrict__ P, float* __restrict__ ws)
{
    __shared__ float lam[16];
    __shared__ float Pl[256];
    __shared__ float core[256];
    __shared__ float tmp[256];
    __shared__ float aug[16 * 96];
    __shared__ float TL[256];
    __shared__ float Tn[256];
    const int tid = threadIdx.x;
    const int i = tid >> 4, j = tid & 15;

    if (tid < 16) lam[tid] = -expf(log_lambda[tid]);
    Pl[tid] = P[tid];
    __syncthreads();

    core[tid] = ((i == j) ? lam[i] : 0.0f) + p[i] * q[j];
    __syncthreads();

    float s = 0.0f;
    #pragma unroll
    for (int k = 0; k < 16; k++) s += Pl[i * 16 + k] * core[k * 16 + j];
    tmp[tid] = s;
    __syncthreads();

    float ac = 0.0f;
    #pragma unroll
    for (int k = 0; k < 16; k++) ac += tmp[i * 16 + k] * Pl[j * 16 + k];
    core[tid] = ac;                 // core now holds A_c
    __syncthreads();

    // augmented matrix [ I-0.5A | I+0.5A | DT*B ]  (16 x 96)
    for (int cc = tid; cc < 1536; cc += 256) {
        int ri = cc / 96, cj = cc % 96;
        float v;
        if (cj < 16)       v = ((ri == cj) ? 1.0f : 0.0f) - 0.5f * core[ri * 16 + cj];
        else if (cj < 32)  v = ((ri == cj - 16) ? 1.0f : 0.0f) + 0.5f * core[ri * 16 + (cj - 16)];
        else               v = 1.0f /*DT*/ * B[ri * 64 + (cj - 32)];
        aug[cc] = v;
    }
    __syncthreads();

    // Gauss-Jordan (no pivoting: M is strongly diagonally dominant here)
    for (int k = 0; k < 16; k++) {
        float piv = aug[k * 96 + k];
        __syncthreads();
        float pinv = 1.0f / piv;
        if (tid < 96) aug[k * 96 + tid] *= pinv;
        __syncthreads();
        // eliminate rows i != k; skip column k (never read again) -> no hazard
        for (int cc = tid; cc < 1440; cc += 256) {
            int ip = cc / 96, cj = cc % 96;
            int ri = ip + (ip >= k ? 1 : 0);
            if (cj != k) aug[ri * 96 + cj] -= aug[ri * 96 + k] * aug[k * 96 + cj];
        }
        __syncthreads();
    }

    // extract A_d and B_d
    ws[WS_AD + tid] = aug[i * 96 + 16 + j];
    TL[tid]         = aug[i * 96 + 16 + j];
    for (int cc = tid; cc < 1024; cc += 256) {
        int ri = cc >> 6, d = cc & 63;
        ws[WS_BD + cc] = aug[ri * 96 + 32 + d];
    }
    // transposes for WMMA B-fragments
    for (int cc = tid; cc < 1024; cc += 256) {
        int kk = cc >> 6, o = cc & 63;
        ws[WS_CT + cc] = C[o * 16 + kk];
    }
    for (int cc = tid; cc < 4096; cc += 256) {
        int kk = cc >> 6, o = cc & 63;
        ws[WS_DT + cc] = D[o * 64 + kk];
    }
    __syncthreads();

    // T = A_d^CHUNK by repeated squaring
    for (int sq = 0; sq < LOG2_CHUNK; sq++) {
        float acc = 0.0f;
        #pragma unroll
        for (int k = 0; k < 16; k++) acc += TL[i * 16 + k] * TL[k * 16 + j];
        Tn[tid] = acc;
        __syncthreads();
        TL[tid] = Tn[tid];
        __syncthreads();
    }
    ws[WS_T + tid] = TL[tid];
}

// ---------------------------------------------------------------------------
// K2: per-chunk: U_t = B_d x_t (stored for K4) and local recurrence from
// zero init -> local final state f[c][b][:].  thread = (batch, state).
// X tiles staged global->LDS with double-buffered async copies (ASYNCcnt):
// step t+1's 8KB tile is in flight while step t's projection runs.
// ---------------------------------------------------------------------------
__global__ __launch_bounds__(512) void ssm_chunk_scan(
    const float* __restrict__ x, float* __restrict__ ws)
{
    __shared__ float BdL[1024];
    __shared__ float AdL[256];
    __shared__ float xst[2][BATCH * DIN];   // 2 x 2048 (double buffer)
    __shared__ float hbuf[512];
    const int tid = threadIdx.x;
    const int c = blockIdx.x;
    for (int idx = tid; idx < 1024; idx += 512) BdL[idx] = ws[WS_BD + idx];
    if (tid < 256) AdL[tid] = ws[WS_AD + tid];
    const int b = tid >> 4, i = tid & 15;

    // each thread owns 16B (4 floats) of the per-step X tile
    const int f0 = tid * 4;
    const int xb = f0 >> 6;           // batch row of this thread's 16B
    const int xd = f0 & 63;           // column offset within the row
    const unsigned lds0 = (unsigned)(uintptr_t)(&xst[0][f0]);
    const unsigned lds1 = (unsigned)(uintptr_t)(&xst[1][f0]);

    float h = 0.0f;

    // prologue: kick off async copy of tile t=0 into buffer 0
    {
        const float* g = x + ((size_t)xb * SEQ + (size_t)c * CHUNK + 0) * DIN + xd;
        asm volatile("global_load_async_to_lds_b128 %0, %1, off"
                     :: "v"(lds0), "v"(g) : "memory");
    }
    __syncthreads();

    for (int t = 0; t < CHUNK; t++) {
        const int tg  = c * CHUNK + t;
        const int cur = t & 1;
        // prefetch next tile into the other buffer, then wait so that only
        // the prefetch remains outstanding (this wave's tile t is complete)
        if (t + 1 < CHUNK) {
            const unsigned ldsn = (t & 1) ? lds0 : lds1;
            const float* g = x + ((size_t)xb * SEQ + tg + 1) * DIN + xd;
            asm volatile("global_load_async_to_lds_b128 %0, %1, off"
                         :: "v"(ldsn), "v"(g) : "memory");
            asm volatile("s_wait_asynccnt 0x1" ::: "memory");
        } else {
            asm volatile("s_wait_asynccnt 0x0" ::: "memory");
        }
        hbuf[tid] = h;
        __syncthreads();   // all waves' async writes to xst[cur] now visible

        float u = 0.0f;
        #pragma unroll
        for (int d = 0; d < DIN; d++) u += BdL[i * DIN + d] * xst[cur][b * DIN + d];
        float hn = u;
        #pragma unroll
        for (int k = 0; k < 16; k++) hn += AdL[i * 16 + k] * hbuf[b * 16 + k];
        ws[WS_U + ((size_t)b * SEQ + tg) * N16 + i] = u;
        __syncthreads();
        h = hn;
    }
    ws[WS_F + (size_t)c * (BATCH * N16) + tid] = h;
}

// ---------------------------------------------------------------------------
// K3: serial carry scan over chunks: Hin[c+1] = T*Hin[c] + f[c], Hin[0]=0.
// ---------------------------------------------------------------------------
__global__ __launch_bounds__(512) void ssm_carry(float* __restrict__ ws)
{
    __shared__ float TL[256];
    __shared__ float hbuf[512];
    const int tid = threadIdx.x;
    if (tid < 256) TL[tid] = ws[WS_T + tid];
    const int b = tid >> 4, i = tid & 15;
    float h = 0.0f;
    __syncthreads();
    for (int c = 0; c < NCHUNK; c++) {
        ws[WS_HIN + (size_t)c * (BATCH * N16) + tid] = h;
        hbuf[tid] = h;
        __syncthreads();
        float hn = ws[WS_F + (size_t)c * (BATCH * N16) + tid];
        #pragma unroll
        for (int k = 0; k < 16; k++) hn += TL[i * 16 + k] * hbuf[b * 16 + k];
        __syncthreads();
        h = hn;
    }
}

// ---------------------------------------------------------------------------
// K4: per-chunk: recurrence from corrected Hin -> hs in LDS, then fused
// WMMA fp32 GEMM:  Y[r][o] = Hs[r][:]·Ct[:][o] + X[r][:]·Dt[:][o]
// with r = b*CHUNK + t  (512 rows = 32 M-tiles, 16 waves -> 2 tiles/wave).
// ---------------------------------------------------------------------------
__global__ __launch_bounds__(512) void ssm_output(
    const float* __restrict__ x, const float* __restrict__ ws,
    float* __restrict__ y)
{
    __shared__ float AdL[256];
    __shared__ float CtL[1024];
    __shared__ float DtL[4096];
    __shared__ float hsL[BATCH * CHUNK * N16];  // 8192 floats
    __shared__ float hbuf[512];
    const int tid = threadIdx.x;
    const int c = blockIdx.x;
    if (tid < 256) AdL[tid] = ws[WS_AD + tid];
    for (int idx = tid; idx < 1024; idx += 512) CtL[idx] = ws[WS_CT + idx];
    for (int idx = tid; idx < 4096; idx += 512) DtL[idx] = ws[WS_DT + idx];
    const int b = tid >> 4, i = tid & 15;
    float h = ws[WS_HIN + (size_t)c * (BATCH * N16) + tid];
    __syncthreads();

    // corrected local recurrence; hs rows laid out r = b*CHUNK + t
    for (int t = 0; t < CHUNK; t++) {
        hbuf[tid] = h;
        __syncthreads();
        float hn = ws[WS_U + ((size_t)b * SEQ + c * CHUNK + t) * N16 + i];
        #pragma unroll
        for (int k = 0; k < 16; k++) hn += AdL[i * 16 + k] * hbuf[b * 16 + k];
        hsL[(b * CHUNK + t) * N16 + i] = hn;
        __syncthreads();
        h = hn;
    }

    // WMMA GEMM phase (uniform control flow, EXEC all ones)
    const int lane = tid & 31;
    const int wv   = tid >> 5;                 // 16 wave32 waves
    const int half = lane >> 4;                // 0: lanes 0-15, 1: lanes 16-31
    const int lm   = lane & 15;

    for (int mt = wv; mt < (BATCH * CHUNK) / 16; mt += 16) {
        const int mBase = mt * 16;
        v8f acc0 = {}, acc1 = {}, acc2 = {}, acc3 = {};

        // Hs (R x 16) @ Ct (16 x 64) : 4 K-steps of K=4
        #pragma unroll
        for (int kk = 0; kk < 4; kk++) {
            int k0 = kk * 4 + 2 * half;        // A layout: VGPR0 K=0/2, VGPR1 K=1/3
            v2f a;
            a.x = hsL[(mBase + lm) * 16 + k0];
            a.y = hsL[(mBase + lm) * 16 + k0 + 1];
            v2f b0, b1, b2, b3;
            b0.x = CtL[k0 * 64 + 0  + lm]; b0.y = CtL[(k0 + 1) * 64 + 0  + lm];
            b1.x = CtL[k0 * 64 + 16 + lm]; b1.y = CtL[(k0 + 1) * 64 + 16 + lm];
            b2.x = CtL[k0 * 64 + 32 + lm]; b2.y = CtL[(k0 + 1) * 64 + 32 + lm];
            b3.x = CtL[k0 * 64 + 48 + lm]; b3.y = CtL[(k0 + 1) * 64 + 48 + lm];
            acc0 = __builtin_amdgcn_wmma_f32_16x16x4_f32(false, a, false, b0, (short)0, acc0, false, false);
            acc1 = __builtin_amdgcn_wmma_f32_16x16x4_f32(false, a, false, b1, (short)0, acc1, false, false);
            acc2 = __builtin_amdgcn_wmma_f32_16x16x4_f32(false, a, false, b2, (short)0, acc2, false, false);
            acc3 = __builtin_amdgcn_wmma_f32_16x16x4_f32(false, a, false, b3, (short)0, acc3, false, false);
        }

        // X (R x 64) @ Dt (64 x 64) : 16 K-steps of K=4, A-frags from global x
        const int r  = mBase + lm;
        const int bb = r >> 4;                 // r / CHUNK
        const int tt = r & (CHUNK - 1);
        const float* xrow = x + ((size_t)bb * SEQ + c * CHUNK + tt) * DIN;
        #pragma unroll
        for (int kk = 0; kk < 16; kk++) {
            int k0 = kk * 4 + 2 * half;
            v2f a;
            a.x = xrow[k0];
            a.y = xrow[k0 + 1];
            v2f b0, b1, b2, b3;
            b0.x = DtL[k0 * 64 + 0  + lm]; b0.y = DtL[(k0 + 1) * 64 + 0  + lm];
            b1.x = DtL[k0 * 64 + 16 + lm]; b1.y = DtL[(k0 + 1) * 64 + 16 + lm];
            b2.x = DtL[k0 * 64 + 32 + lm]; b2.y = DtL[(k0 + 1) * 64 + 32 + lm];
            b3.x = DtL[k0 * 64 + 48 + lm]; b3.y = DtL[(k0 + 1) * 64 + 48 + lm];
            acc0 = __builtin_amdgcn_wmma_f32_16x16x4_f32(false, a, false, b0, (short)0, acc0, false, false);
            acc1 = __builtin_amdgcn_wmma_f32_16x16x4_f32(false, a, false, b1, (short)0, acc1, false, false);
            acc2 = __builtin_amdgcn_wmma_f32_16x16x4_f32(false, a, false, b2, (short)0, acc2, false, false);
            acc3 = __builtin_amdgcn_wmma_f32_16x16x4_f32(false, a, false, b3, (short)0, acc3, false, false);
        }

        // store: C/D layout — VGPR v holds rows v (lanes 0-15) / v+8 (lanes 16-31)
        #pragma unroll
        for (int v = 0; v < 8; v++) {
            const int r2  = mBase + v + 8 * half;
            const int bb2 = r2 >> 4;
            const int tt2 = r2 & (CHUNK - 1);
            float* yrow = y + ((size_t)bb2 * SEQ + c * CHUNK + tt2) * DOUT;
            yrow[0  + lm] = acc0[v];
            yrow[16 + lm] = acc1[v];
            yrow[32 + lm] = acc2[v];
            yrow[48 + lm] = acc3[v];
        }
    }
}

// ---------------------------------------------------------------------------
extern "C" void kernel_launch(void* const* d_in, const int* in_sizes, int n_in,
                              void* d_out, int out_size, void* d_ws, size_t ws_size,
                              hipStream_t stream) {
    (void)in_sizes; (void)n_in; (void)out_size; (void)ws_size;
    const float* x  = (const float*)d_in[0];
    const float* ll = (const float*)d_in[1];
    const float* p  = (const float*)d_in[2];
    const float* q  = (const float*)d_in[3];
    const float* B  = (const float*)d_in[4];
    const float* C  = (const float*)d_in[5];
    const float* D  = (const float*)d_in[6];
    const float* P  = (const float*)d_in[7];
    float* ws = (float*)d_ws;
    float* y  = (float*)d_out;

    ssm_setup<<<1, 256, 0, stream>>>(ll, p, q, B, C, D, P, ws);
    ssm_chunk_scan<<<NCHUNK, 512, 0, stream>>>(x, ws);
    ssm_carry<<<1, 512, 0, stream>>>(ws);
    ssm_output<<<NCHUNK, 512, 0, stream>>>(x, ws, y);
}